// MyGnn_55147380080805
// MI455X (gfx1250) — compile-verified
//
#include <hip/hip_runtime.h>

#define NN   50000
#define NE   800000
#define ET   (NE + NN)       // 850000, divisible by 16
#define FEATD 29
#define POSD  3
#define HD    64
#define OUTD  32

typedef __attribute__((ext_vector_type(2))) float v2f;
typedef __attribute__((ext_vector_type(8))) float v8f;

// ---------------- WMMA fp32 16x16x4 helper -------------------------------
__device__ __forceinline__ v8f wmma4(v2f a, v2f b, v8f c) {
  // 8 args: (neg_a, A, neg_b, B, c_mod, C, reuse_a, reuse_b)
  return __builtin_amdgcn_wmma_f32_16x16x4_f32(false, a, false, b, (short)0, c,
                                               false, false);
}

// order-preserving int encoding of float (involution) for atomic max
__device__ __forceinline__ int enc_f(float f) {
  int i = __float_as_int(f);
  return i >= 0 ? i : (i ^ 0x7fffffff);
}
__device__ __forceinline__ float dec_f(int i) {
  return __int_as_float(i >= 0 ? i : (i ^ 0x7fffffff));
}
#define ENC_NEG_INF ((int)0x807fffffu)

__device__ __forceinline__ void edge_sd(const int* __restrict__ ei, long e,
                                        int& s, int& d) {
  if (e < NE) { s = ei[e]; d = ei[NE + e]; }
  else        { s = d = (int)(e - NE); }
}

// prefetch nfloats of read-only data (weights) into cache: global_prefetch_b8
__device__ __forceinline__ void prefetch_ro(const float* p, int nfloats, int lane) {
  for (int i = lane * 32; i < nfloats; i += 32 * 32)   // one lane per 128B line
    __builtin_prefetch(p + i, 0, 3);
}

// D[16 x NOUT] = act(A[16 x K] * W[K x NOUT] + bias), one wave, via WMMA.
// A,O in LDS (row-major, strides LDA/LDO). W,bias in global.
template <int K, int NOUT, int LDA, int LDO, bool RELU, bool BIAS>
__device__ __forceinline__ void wave_mlp_tile(const float* A,
                                              const float* __restrict__ W,
                                              const float* __restrict__ bias,
                                              float* O, int lane) {
  const int half = lane >> 4;      // 0: lanes 0-15, 1: lanes 16-31
  const int nlo  = lane & 15;
#pragma unroll
  for (int n0 = 0; n0 < NOUT; n0 += 16) {
    const float bv = BIAS ? bias[n0 + nlo] : 0.f;
    v8f c = {};
#pragma unroll
    for (int k0 = 0; k0 < K; k0 += 4) {
      const int ka = k0 + 2 * half;
      v2f av, wv;
      av[0] = A[nlo * LDA + ka];
      av[1] = A[nlo * LDA + ka + 1];
      wv[0] = W[ka * NOUT + n0 + nlo];
      wv[1] = W[(ka + 1) * NOUT + n0 + nlo];
      c = wmma4(av, wv, c);
    }
#pragma unroll
    for (int r = 0; r < 8; ++r) {
      float v = c[r] + bv;
      if (RELU) v = fmaxf(v, 0.f);
      O[(r + 8 * half) * LDO + n0 + nlo] = v;
    }
  }
}

// ---------------- PointNetConv message MLP + segment-max -----------------
__global__ __launch_bounds__(32)
void point_conv_kernel(const float* __restrict__ x, const float* __restrict__ pos,
                       const int* __restrict__ ei,
                       const float* __restrict__ W1, const float* __restrict__ b1,
                       const float* __restrict__ W2, const float* __restrict__ b2,
                       int* __restrict__ agg_enc) {
  __shared__ float sIn[16 * 33];    // 16 msgs x 32 feats (padded)
  __shared__ float sMid[16 * 65];   // layer-1 output 16 x 64 (padded)
  __shared__ int   sDst[16];
  const int lane = threadIdx.x;
  const long e0 = (long)blockIdx.x * 16;

  prefetch_ro(W1, 32 * HD, lane);
  prefetch_ro(W2, HD * HD, lane);

  for (int i = lane; i < 16 * 32; i += 32) {
    const int r = i >> 5, c = i & 31;
    int src, dst;
    edge_sd(ei, e0 + r, src, dst);
    float v;
    if (c < FEATD) v = x[(long)src * FEATD + c];
    else v = pos[(long)src * POSD + (c - FEATD)] - pos[(long)dst * POSD + (c - FEATD)];
    sIn[r * 33 + c] = v;
    if (c == 0) sDst[r] = dst;
  }
  __syncthreads();
  wave_mlp_tile<32, HD, 33, 65, true, true>(sIn, W1, b1, sMid, lane);  // ReLU
  __syncthreads();

  const int half = lane >> 4, nlo = lane & 15;
#pragma unroll
  for (int n0 = 0; n0 < HD; n0 += 16) {
    const float bv = b2[n0 + nlo];
    v8f c = {};
#pragma unroll
    for (int k0 = 0; k0 < HD; k0 += 4) {
      const int ka = k0 + 2 * half;
      v2f av, wv;
      av[0] = sMid[nlo * 65 + ka];
      av[1] = sMid[nlo * 65 + ka + 1];
      wv[0] = W2[ka * HD + n0 + nlo];
      wv[1] = W2[(ka + 1) * HD + n0 + nlo];
      c = wmma4(av, wv, c);
    }
#pragma unroll
    for (int r = 0; r < 8; ++r) {
      const int m = r + 8 * half;
      atomicMax(&agg_enc[(long)sDst[m] * HD + n0 + nlo], enc_f(c[r] + bv));
    }
  }
}

// ---------------- PointNetConv global MLP (64->32->128->64) --------------
__global__ __launch_bounds__(32)
void glob_mlp_kernel(const int* __restrict__ agg_enc,
                     const float* __restrict__ W1, const float* __restrict__ b1,
                     const float* __restrict__ W2, const float* __restrict__ b2,
                     const float* __restrict__ W3, const float* __restrict__ b3,
                     float* __restrict__ hout) {
  __shared__ float sA[16 * 129];
  __shared__ float sB[16 * 129];
  const int lane = threadIdx.x;
  const long r0 = (long)blockIdx.x * 16;

  prefetch_ro(W1, HD * 32, lane);
  prefetch_ro(W2, 32 * 128, lane);
  prefetch_ro(W3, 128 * HD, lane);

  for (int i = lane; i < 16 * HD; i += 32) {
    const int r = i >> 6, c = i & 63;
    sA[r * 65 + c] = dec_f(agg_enc[(r0 + r) * HD + c]);
  }
  __syncthreads();
  wave_mlp_tile<64, 32, 65, 33, true, true>(sA, W1, b1, sB, lane);
  __syncthreads();
  wave_mlp_tile<32, 128, 33, 129, true, true>(sB, W2, b2, sA, lane);
  __syncthreads();
  wave_mlp_tile<128, 64, 129, 65, false, true>(sA, W3, b3, sB, lane);
  __syncthreads();
  for (int i = lane; i < 16 * HD; i += 32) {
    const int r = i >> 6, c = i & 63;
    hout[(r0 + r) * HD + c] = sB[r * 65 + c];
  }
}

// ---------------- h = feat @ W (+ optional GAT attention dots) -----------
template <int NOUT>
__global__ __launch_bounds__(32)
void node_gemm_kernel(const float* __restrict__ feat, const float* __restrict__ W,
                      float* __restrict__ hout,
                      const float* __restrict__ attS, const float* __restrict__ attD,
                      float* __restrict__ asOut, float* __restrict__ adOut) {
  __shared__ float sIn[16 * 65];
  __shared__ float sOut[16 * 65];
  const int lane = threadIdx.x;
  const long r0 = (long)blockIdx.x * 16;

  prefetch_ro(W, HD * NOUT, lane);

  for (int i = lane; i < 16 * HD; i += 32) {
    const int r = i >> 6, c = i & 63;
    sIn[r * 65 + c] = feat[(r0 + r) * HD + c];
  }
  __syncthreads();
  wave_mlp_tile<HD, NOUT, 65, 65, false, false>(sIn, W, nullptr, sOut, lane);
  __syncthreads();
  for (int i = lane; i < 16 * NOUT; i += 32) {
    const int r = i / NOUT, c = i % NOUT;
    hout[(r0 + r) * NOUT + c] = sOut[r * 65 + c];
  }
  if (attS) {
    const int row = lane & 15;
    const float* av = (lane >= 16) ? attD : attS;
    float s = 0.f;
#pragma unroll 8
    for (int k = 0; k < HD; ++k) s += sOut[row * 65 + k] * av[k];
    if (lane >= 16) adOut[r0 + row] = s;
    else            asOut[r0 + row] = s;
  }
}

// ---------------- GAT edge passes ----------------------------------------
__global__ void gat_edge1(const int* __restrict__ ei, const float* __restrict__ as_,
                          const float* __restrict__ ad_, float* __restrict__ esc,
                          int* __restrict__ menc) {
  const long e = (long)blockIdx.x * blockDim.x + threadIdx.x;
  if (e >= ET) return;
  int s, d; edge_sd(ei, e, s, d);
  float v = as_[s] + ad_[d];
  v = v > 0.f ? v : 0.2f * v;             // leaky_relu(0.2)
  esc[e] = v;
  atomicMax(&menc[d], enc_f(v));
}

__global__ void gat_edge2(const int* __restrict__ ei, const int* __restrict__ menc,
                          float* __restrict__ esc, float* __restrict__ ssum) {
  const long e = (long)blockIdx.x * blockDim.x + threadIdx.x;
  if (e >= ET) return;
  int s, d; edge_sd(ei, e, s, d);
  const float w = expf(esc[e] - dec_f(menc[d]));
  esc[e] = w;
  atomicAdd(&ssum[d], w);
}

__global__ void gat_edge3(const int* __restrict__ ei, const float* __restrict__ esc,
                          const float* __restrict__ ssum, const float* __restrict__ h,
                          float* __restrict__ acc) {
  const long idx = (long)blockIdx.x * blockDim.x + threadIdx.x;
  if (idx >= (long)ET * HD) return;
  const long e = idx >> 6;
  const int  f = (int)(idx & 63);
  int s, d; edge_sd(ei, e, s, d);
  const float alpha = esc[e] / ssum[d];
  atomicAdd(&acc[(long)d * HD + f], alpha * h[(long)s * HD + f]);
}

// ---------------- GCN helpers --------------------------------------------
__global__ void deg_kernel(const int* __restrict__ ei, float* __restrict__ deg) {
  const long e = (long)blockIdx.x * blockDim.x + threadIdx.x;
  if (e >= ET) return;
  int s, d; edge_sd(ei, e, s, d);
  atomicAdd(&deg[d], 1.0f);
}
__global__ void rsqrt_kernel(const float* __restrict__ deg, float* __restrict__ dinv,
                             int n) {
  const int i = blockIdx.x * blockDim.x + threadIdx.x;
  if (i < n) dinv[i] = rsqrtf(deg[i]);
}
__global__ void gcn_edge(const int* __restrict__ ei, const float* __restrict__ dinv,
                         const float* __restrict__ h, float* __restrict__ acc,
                         int fshift) {
  const long idx = (long)blockIdx.x * blockDim.x + threadIdx.x;
  if (idx >= ((long)ET << fshift)) return;
  const long e = idx >> fshift;
  const int  f = (int)(idx & ((1 << fshift) - 1));
  int s, d; edge_sd(ei, e, s, d);
  const float nrm = dinv[s] * dinv[d];
  atomicAdd(&acc[((long)d << fshift) + f], nrm * h[((long)s << fshift) + f]);
}

// ---------------- misc ----------------------------------------------------
__global__ void fill_i32(int* __restrict__ p, long n, int v) {
  const long i = (long)blockIdx.x * blockDim.x + threadIdx.x;
  if (i < n) p[i] = v;
}
__global__ void bias_act(float* __restrict__ a, const float* __restrict__ b,
                         int mask, int relu, long n) {
  const long i = (long)blockIdx.x * blockDim.x + threadIdx.x;
  if (i >= n) return;
  float v = a[i] + b[(int)(i & mask)];
  if (relu) v = fmaxf(v, 0.f);
  a[i] = v;
}

// ---------------- host ----------------------------------------------------
extern "C" void kernel_launch(void* const* d_in, const int* in_sizes, int n_in,
                              void* d_out, int out_size, void* d_ws, size_t ws_size,
                              hipStream_t stream) {
  (void)n_in; (void)out_size; (void)ws_size;
  auto F = [&](int i) { return (const float*)d_in[i]; };
  const float *x, *pos;
  const int* ei;
  const float *pl0w, *pl0b, *pl1w, *pl1b;
  const float *pg0w, *pg0b, *pg1w, *pg1b, *pg2w, *pg2b;
  const float *gw[2], *gas[2], *gad[2], *gb[2];
  const float *cw[2], *cb[2], *ow, *ob;

  if (in_sizes[0] == NN * FEATD) {      // dict insertion-order flattening
    x = F(0); pos = F(1); ei = (const int*)d_in[2];
    pl0w = F(3);  pl0b = F(4);  pl1w = F(5);  pl1b = F(6);
    pg0w = F(7);  pg0b = F(8);  pg1w = F(9);  pg1b = F(10);
    pg2w = F(11); pg2b = F(12);
    gw[0] = F(13); gas[0] = F(14); gad[0] = F(15); gb[0] = F(16);
    gw[1] = F(17); gas[1] = F(18); gad[1] = F(19); gb[1] = F(20);
    cw[0] = F(21); cb[0] = F(22); cw[1] = F(23); cb[1] = F(24);
    ow = F(25); ob = F(26);
  } else {                              // jax pytree sorted-key flattening
    ei = (const int*)d_in[0];
    gad[0] = F(1); gas[0] = F(2); gb[0] = F(3); gw[0] = F(4);
    gad[1] = F(5); gas[1] = F(6); gb[1] = F(7); gw[1] = F(8);
    cb[0] = F(9);  cw[0] = F(10); cb[1] = F(11); cw[1] = F(12);
    ob = F(13); ow = F(14);
    pg0b = F(15); pg0w = F(16); pg1b = F(17); pg1w = F(18);
    pg2b = F(19); pg2w = F(20);
    pl0b = F(21); pl0w = F(22); pl1b = F(23); pl1w = F(24);
    pos = F(25); x = F(26);
  }

  char* w = (char*)d_ws;
  float* h0 = (float*)w;  w += (size_t)NN * HD * 4;
  float* h1 = (float*)w;  w += (size_t)NN * HD * 4;
  float* h2 = (float*)w;  w += (size_t)NN * HD * 4;
  int*   agg = (int*)h2;                 // aliases h2 (agg dead before h2 use)
  float* as_ = (float*)w; w += (size_t)NN * 4;
  float* ad_ = (float*)w; w += (size_t)NN * 4;
  float* ss  = (float*)w; w += (size_t)NN * 4;
  int*   men = (int*)w;   w += (size_t)NN * 4;
  float* div_ = (float*)w; w += (size_t)NN * 4;
  float* esc = (float*)w; w += (size_t)ET * 4;
  float* out = (float*)d_out;

  const int TB = 256;
  const int gN   = (NN + TB - 1) / TB;
  const int gNH  = (NN * HD + TB - 1) / TB;
  const int gE   = (ET + TB - 1) / TB;
  const int gEH  = (int)(((long)ET * HD + TB - 1) / TB);
  const int gEO  = (int)(((long)ET * OUTD + TB - 1) / TB);
  const int gNO  = (NN * OUTD + TB - 1) / TB;

  // ---- PointNetConv ----
  fill_i32<<<gNH, TB, 0, stream>>>(agg, (long)NN * HD, ENC_NEG_INF);
  point_conv_kernel<<<ET / 16, 32, 0, stream>>>(x, pos, ei, pl0w, pl0b, pl1w, pl1b, agg);
  glob_mlp_kernel<<<NN / 16, 32, 0, stream>>>(agg, pg0w, pg0b, pg1w, pg1b, pg2w, pg2b, h0);

  float* featB[2] = {h0, h2};
  float* accB[2]  = {h2, h0};

  // ---- 2x GATConv (+ReLU) ----
  for (int l = 0; l < 2; ++l) {
    node_gemm_kernel<HD><<<NN / 16, 32, 0, stream>>>(featB[l], gw[l], h1,
                                                     gas[l], gad[l], as_, ad_);
    fill_i32<<<gN, TB, 0, stream>>>(men, NN, ENC_NEG_INF);
    fill_i32<<<gN, TB, 0, stream>>>((int*)ss, NN, 0);
    fill_i32<<<gNH, TB, 0, stream>>>((int*)accB[l], (long)NN * HD, 0);
    gat_edge1<<<gE, TB, 0, stream>>>(ei, as_, ad_, esc, men);
    gat_edge2<<<gE, TB, 0, stream>>>(ei, men, esc, ss);
    gat_edge3<<<gEH, TB, 0, stream>>>(ei, esc, ss, h1, accB[l]);
    bias_act<<<gNH, TB, 0, stream>>>(accB[l], gb[l], HD - 1, 1, (long)NN * HD);
  }

  // ---- degree norm (shared by all GCN layers) ----
  fill_i32<<<gN, TB, 0, stream>>>((int*)ss, NN, 0);
  deg_kernel<<<gE, TB, 0, stream>>>(ei, ss);
  rsqrt_kernel<<<gN, TB, 0, stream>>>(ss, div_, NN);

  // ---- 2x GCNConv (+ReLU) ----
  for (int l = 0; l < 2; ++l) {
    node_gemm_kernel<HD><<<NN / 16, 32, 0, stream>>>(featB[l], cw[l], h1,
                                                     nullptr, nullptr, nullptr, nullptr);
    fill_i32<<<gNH, TB, 0, stream>>>((int*)accB[l], (long)NN * HD, 0);
    gcn_edge<<<gEH, TB, 0, stream>>>(ei, div_, h1, accB[l], 6);
    bias_act<<<gNH, TB, 0, stream>>>(accB[l], cb[l], HD - 1, 1, (long)NN * HD);
  }

  // ---- output GCNConv (64 -> 32, no ReLU) ----
  node_gemm_kernel<OUTD><<<NN / 16, 32, 0, stream>>>(h0, ow, h1,
                                                     nullptr, nullptr, nullptr, nullptr);
  fill_i32<<<gNO, TB, 0, stream>>>((int*)out, (long)NN * OUTD, 0);
  gcn_edge<<<gEO, TB, 0, stream>>>(ei, div_, h1, out, 5);
  bias_act<<<gNO, TB, 0, stream>>>(out, ob, OUTD - 1, 0, (long)NN * OUTD);
}